// Linear_35493609734470
// MI455X (gfx1250) — compile-verified
//
#include <hip/hip_runtime.h>

typedef __attribute__((ext_vector_type(2))) float v2f;
typedef __attribute__((ext_vector_type(8))) float v8f;

#define U_BINS 1024
#define RDIM 8
#define CDIM 8
#define GEMV_THREADS 512
#define GEMV_WAVES (GEMV_THREADS / 32)

// ---------- Phase 1: zero histogram + cursors ----------
__global__ void zero_kernel(int* counts, int* cursor) {
    int i = blockIdx.x * blockDim.x + threadIdx.x;
    if (i < U_BINS) { counts[i] = 0; cursor[i] = 0; }
}

// ---------- Phase 2: histogram of weight ids ----------
__global__ void hist_kernel(const int* __restrict__ widx, int n, int* counts) {
    int i = blockIdx.x * blockDim.x + threadIdx.x;
    if (i < n) atomicAdd(&counts[widx[i]], 1);
}

// ---------- Phase 3: exclusive scan over 1024 bins (one block) ----------
__global__ void __launch_bounds__(U_BINS)
scan_kernel(const int* __restrict__ counts, int* __restrict__ offsets) {
    __shared__ int sm[U_BINS];
    int t = threadIdx.x;
    sm[t] = counts[t];
    __syncthreads();
    for (int d = 1; d < U_BINS; d <<= 1) {
        int v = (t >= d) ? sm[t - d] : 0;
        __syncthreads();
        sm[t] += v;
        __syncthreads();
    }
    offsets[t + 1] = sm[t];          // inclusive -> shifted = exclusive
    if (t == 0) offsets[0] = 0;
}

// ---------- Phase 4: counting-sort scatter of (neuron id, x-row id) pairs ----------
__global__ void scatter_kernel(const int* __restrict__ widx,
                               const int* __restrict__ input_idx, int n,
                               const int* __restrict__ offsets,
                               int* cursor, int2* __restrict__ pairs) {
    int i = blockIdx.x * blockDim.x + threadIdx.x;
    if (i < n) {
        int u = widx[i];
        int p = atomicAdd(&cursor[u], 1);
        pairs[offsets[u] + p] = make_int2(i, input_idx[i]);
    }
}

// ---------- Phase 5: per-weight batched GEMV via V_WMMA_F32_16X16X4_F32 ----------
// One workgroup per unique weight u. Tile = 16 neurons sharing W_u.
//   D[r][j] = sum_c W_u[r][c] * x_j[c]
// A = W_u (16x4 chunk, rows = r, rows 8..15 zero, LOOP-INVARIANT)
// B = X^T  (4x16 chunk, columns = neurons)
// => lane j holds neuron j's full 8-float result in d[0..7] -> 2x b128 stores.
__global__ void __launch_bounds__(GEMV_THREADS)
wmma_gemv_kernel(const float* __restrict__ values,
                 const float* __restrict__ weights,
                 const int*   __restrict__ offsets,
                 const long long* __restrict__ pairs,   // int2 as one 64-bit load
                 float*       __restrict__ out) {
    const int u    = blockIdx.x;           // unique-weight id
    const int lane = threadIdx.x & 31;
    const int wave = threadIdx.x >> 5;

    const int beg = offsets[u];
    const int cnt = offsets[u + 1] - beg;
    if (cnt == 0) return;                  // block-uniform

    const int j = lane & 15;               // A row (r) / B,D column (neuron slot)
    const int h = lane >> 4;               // K-half (A/B VGPR layout)

    // A = W_u chunks, loop invariant. ISA 16x4 f32 layout:
    // lanes 0-15: VGPR0=K0,VGPR1=K1; lanes 16-31: VGPR0=K2,VGPR1=K3.
    const float* W = weights + (size_t)u * (RDIM * CDIM);
    v2f a0 = {0.f, 0.f}, a1 = {0.f, 0.f};
    if (j < RDIM) {
        a0 = *(const v2f*)(W + j * CDIM + 0 + 2 * h);
        a1 = *(const v2f*)(W + j * CDIM + 4 + 2 * h);
    }

    const int nfull = cnt >> 4;

    // ---- full tiles: zero predication, pair-load software-pipelined ----
    long long pv = 0;                      // low 32 = neuron id, high 32 = x row
    if (nfull > 0) {
        int t0 = (wave < nfull) ? wave : 0;
        pv = pairs[(unsigned)(beg + (t0 << 4) + j)];
    }
    for (int t = wave; t < nfull; t += GEMV_WAVES) {
        const int nid     = (int)(unsigned)pv;
        const unsigned xo = (unsigned)(pv >> 32) * CDIM;
        v2f b0 = *(const v2f*)(values + (xo + 0 + 2 * h));
        v2f b1 = *(const v2f*)(values + (xo + 4 + 2 * h));

        // Prefetch next tile's pair while WMMA + stores drain (uniform cond).
        const int tn = t + GEMV_WAVES;
        if (tn < nfull)
            pv = pairs[(unsigned)(beg + (tn << 4) + j)];

        v8f d = {0.f, 0.f, 0.f, 0.f, 0.f, 0.f, 0.f, 0.f};
        d = __builtin_amdgcn_wmma_f32_16x16x4_f32(false, a0, false, b0,
                                                  (short)0, d, false, false);
        d = __builtin_amdgcn_wmma_f32_16x16x4_f32(false, a1, false, b1,
                                                  (short)0, d, false, false);

        // Lane j (j<16) holds y[nid][0..7] in d[0..7]: 32B contiguous store.
        if (lane < 16) {
            const unsigned oo = (unsigned)nid * RDIM;
            *(float4*)(out + oo)     = make_float4(d[0], d[1], d[2], d[3]);
            *(float4*)(out + oo + 4) = make_float4(d[4], d[5], d[6], d[7]);
        }
    }

    // ---- at most one partial tail tile per u ----
    if ((cnt & 15) != 0 && wave == (nfull & (GEMV_WAVES - 1))) {
        const int tbase = nfull << 4;
        const bool valid = (tbase + j) < cnt;           // single per-lane predicate
        long long tp = 0;
        v2f b0 = {0.f, 0.f}, b1 = {0.f, 0.f};
        if (valid) {
            tp = pairs[(unsigned)(beg + tbase + j)];
            const unsigned xo = (unsigned)(tp >> 32) * CDIM;
            b0 = *(const v2f*)(values + (xo + 0 + 2 * h));
            b1 = *(const v2f*)(values + (xo + 4 + 2 * h));
        }
        v8f d = {0.f, 0.f, 0.f, 0.f, 0.f, 0.f, 0.f, 0.f};
        d = __builtin_amdgcn_wmma_f32_16x16x4_f32(false, a0, false, b0,
                                                  (short)0, d, false, false);
        d = __builtin_amdgcn_wmma_f32_16x16x4_f32(false, a1, false, b1,
                                                  (short)0, d, false, false);
        if (lane < 16 && valid) {
            const unsigned oo = (unsigned)(int)(unsigned)tp * RDIM;
            *(float4*)(out + oo)     = make_float4(d[0], d[1], d[2], d[3]);
            *(float4*)(out + oo + 4) = make_float4(d[4], d[5], d[6], d[7]);
        }
    }
}

// ---------- Fallback: fully coalesced per-lane GEMV (no scratch needed) ----------
__global__ void gemv_fallback(const float* __restrict__ values,
                              const float* __restrict__ weights,
                              const int*   __restrict__ input_idx,
                              const int*   __restrict__ widx,
                              int n, float* __restrict__ out) {
    int i = blockIdx.x * blockDim.x + threadIdx.x;
    if (i >= n) return;
    const float* x = values + (size_t)input_idx[i] * CDIM;
    float4 x0 = *(const float4*)(x);
    float4 x1 = *(const float4*)(x + 4);
    const float* W = weights + (size_t)widx[i] * (RDIM * CDIM);
    float acc[8];
    #pragma unroll
    for (int rr = 0; rr < 8; ++rr) {
        float4 w0 = *(const float4*)(W + rr * CDIM);
        float4 w1 = *(const float4*)(W + rr * CDIM + 4);
        acc[rr] = w0.x * x0.x + w0.y * x0.y + w0.z * x0.z + w0.w * x0.w
                + w1.x * x1.x + w1.y * x1.y + w1.z * x1.z + w1.w * x1.w;
    }
    *(float4*)(out + (size_t)i * RDIM)     = make_float4(acc[0], acc[1], acc[2], acc[3]);
    *(float4*)(out + (size_t)i * RDIM + 4) = make_float4(acc[4], acc[5], acc[6], acc[7]);
}

extern "C" void kernel_launch(void* const* d_in, const int* in_sizes, int n_in,
                              void* d_out, int out_size, void* d_ws, size_t ws_size,
                              hipStream_t stream) {
    const float* values     = (const float*)d_in[0];  // [M, 8]
    const float* weights    = (const float*)d_in[1];  // [1024, 8, 8]
    const int*   input_idx  = (const int*)  d_in[2];  // [N]
    const int*   weight_idx = (const int*)  d_in[3];  // [N]
    float*       out        = (float*)d_out;          // [N, 8]
    const int n = in_sizes[2];

    // Workspace: counts[1024] | cursor[1024] | offsets[1025] | pairs[N] (int2)
    size_t head = (size_t)(U_BINS + U_BINS + U_BINS + 1) * sizeof(int);
    size_t need = head + 8 /*alignment slack*/ + (size_t)n * sizeof(int2);

    if (ws_size >= need) {
        int* counts  = (int*)d_ws;
        int* cursor  = counts + U_BINS;
        int* offsets = cursor + U_BINS;
        size_t poff = (head + 7) & ~(size_t)7;
        int2* pairs  = (int2*)((char*)d_ws + poff);

        zero_kernel<<<(U_BINS + 255) / 256, 256, 0, stream>>>(counts, cursor);
        hist_kernel<<<(n + 255) / 256, 256, 0, stream>>>(weight_idx, n, counts);
        scan_kernel<<<1, U_BINS, 0, stream>>>(counts, offsets);
        scatter_kernel<<<(n + 255) / 256, 256, 0, stream>>>(weight_idx, input_idx,
                                                            n, offsets, cursor, pairs);
        wmma_gemv_kernel<<<U_BINS, GEMV_THREADS, 0, stream>>>(
            values, weights, offsets, (const long long*)pairs, out);
    } else {
        gemv_fallback<<<(n + 255) / 256, 256, 0, stream>>>(values, weights,
                                                           input_idx, weight_idx,
                                                           n, out);
    }
}